// RandomBitFlipFI_62697932587356
// MI455X (gfx1250) — compile-verified
//
#include <hip/hip_runtime.h>
#include <hip/hip_bf16.h>
#include <stdint.h>

// ---------------------------------------------------------------------------
// RandomBitFlipFI for MI455X (gfx1250).
// Memory-bound streaming map: out[e] = bitcast(x[e]) ^ mask(e), where mask(e)
// flips one random bit for ~1/16 of elements (hash-derived, deterministic).
// Roofline: 512 MB traffic / 23.3 TB/s ~= 22 us; hash trimmed to ~9 VALU/elem
// so compute (with VOPD dual-issue) stays under the memory bound.
// Data path: global_load_async_to_lds_b128 (ASYNCcnt, double-buffered)
//            -> s_wait_asynccnt -> ds_load_b128 -> v_xor -> NT global_store.
// ---------------------------------------------------------------------------

typedef uint32_t v4u __attribute__((ext_vector_type(4)));
typedef int      v4i __attribute__((ext_vector_type(4)));   // builtin's pointee type

#define TPB 256
#define MAX_BLOCKS 8192

#if defined(__gfx1250__) && __has_builtin(__builtin_amdgcn_global_load_async_to_lds_b128)
#define FI_HAVE_ASYNC 1
#else
#define FI_HAVE_ASYNC 0
#endif

// The builtin expects 'int __vector(4)*' (generic AS; compiler infers 1/3).
#define FI_V4IPTR(p) ((v4i*)(void*)(uintptr_t)(p))

__device__ __forceinline__ void fi_wait_async_1() {
#if __has_builtin(__builtin_amdgcn_s_wait_asynccnt)
  __builtin_amdgcn_s_wait_asynccnt(1);
#else
  asm volatile("s_wait_asynccnt 1" ::: "memory");
#endif
}
__device__ __forceinline__ void fi_wait_async_0() {
#if __has_builtin(__builtin_amdgcn_s_wait_asynccnt)
  __builtin_amdgcn_s_wait_asynccnt(0);
#else
  asm volatile("s_wait_asynccnt 0" ::: "memory");
#endif
}

// Deterministic per-element fault mask: P(flip)=1/16, uniform bit position.
// Lean 2-multiply mixer (~9 VALU/elem incl. select+xor): decision bits come
// from the top nibble and the bit position from bits 23..27, both fully
// downstream of the final multiply (best-mixed bits).
__device__ __forceinline__ uint32_t fi_mask(uint32_t e) {
  uint32_t h = (e ^ 0x3243F6A8u) * 0x9E3779B9u;
  h ^= h >> 16;
  h *= 0x85EBCA6Bu;
  uint32_t bit = 1u << ((h >> 23) & 31u);
  return ((h >> 28) == 0u) ? bit : 0u;   // top nibble == 0  =>  P = 1/16
}

__device__ __forceinline__ v4u fi_apply(v4u v, uint32_t e0) {
  v.x ^= fi_mask(e0 + 0u);
  v.y ^= fi_mask(e0 + 1u);
  v.z ^= fi_mask(e0 + 2u);
  v.w ^= fi_mask(e0 + 3u);
  return v;
}

__global__ __launch_bounds__(TPB) void RandomBitFlipFI_stream_kernel(
    const v4u* __restrict__ in, v4u* __restrict__ out, uint32_t n4) {
  const uint32_t tid    = threadIdx.x;
  const uint32_t stride = gridDim.x * TPB;
  uint32_t i = blockIdx.x * TPB + tid;

#if FI_HAVE_ASYNC
  __shared__ v4u lbuf[2][TPB];

  // Prologue: kick off the first tile's async load (memory -> LDS, no VGPRs).
  if (i < n4) {
    __builtin_amdgcn_global_load_async_to_lds_b128(
        FI_V4IPTR(in + i), FI_V4IPTR(&lbuf[0][tid]), 0, 0);
  }
  uint32_t buf = 0;
  for (; i < n4; i += stride, buf ^= 1u) {
    const uint32_t inext = i + stride;
    if (inext < n4) {
      // Issue next tile before consuming current: depth-2 pipeline.
      __builtin_amdgcn_global_load_async_to_lds_b128(
          FI_V4IPTR(in + inext), FI_V4IPTR(&lbuf[buf ^ 1u][tid]), 0, 0);
      fi_wait_async_1();  // in-order completion => tile 'buf' is in LDS
    } else {
      fi_wait_async_0();
    }
    asm volatile("" ::: "memory");          // order DS read after the wait
    v4u v = lbuf[buf][tid];                 // ds_load_b128 (own slot; no barrier)
    v = fi_apply(v, i * 4u);
    __builtin_nontemporal_store(v, out + i);  // TH=NT: don't pollute L2
  }
#else
  // Portable fallback: straight VGPR streaming (still roofline-bound).
  for (; i < n4; i += stride) {
    v4u v = __builtin_nontemporal_load(in + i);
    v = fi_apply(v, i * 4u);
    __builtin_nontemporal_store(v, out + i);
  }
#endif
}

// Scalar tail for n not divisible by 4 (not hit for 64M, kept for generality).
__global__ void RandomBitFlipFI_tail_kernel(const uint32_t* __restrict__ in,
                                            uint32_t* __restrict__ out,
                                            uint32_t start, uint32_t n) {
  uint32_t e = start + blockIdx.x * blockDim.x + threadIdx.x;
  if (e < n) out[e] = in[e] ^ fi_mask(e);
}

extern "C" void kernel_launch(void* const* d_in, const int* in_sizes, int n_in,
                              void* d_out, int out_size, void* d_ws, size_t ws_size,
                              hipStream_t stream) {
  (void)n_in; (void)out_size; (void)d_ws; (void)ws_size;
  const uint32_t n  = (uint32_t)in_sizes[0];   // 67,108,864 fp32 elements
  const uint32_t n4 = n >> 2;

  const v4u* in = (const v4u*)d_in[0];
  v4u*      out = (v4u*)d_out;

  uint32_t blocks = (n4 + TPB - 1) / TPB;
  if (blocks > MAX_BLOCKS) blocks = MAX_BLOCKS;
  if (blocks == 0) blocks = 1;

  RandomBitFlipFI_stream_kernel<<<dim3(blocks), dim3(TPB), 0, stream>>>(in, out, n4);

  if (n & 3u) {
    const uint32_t start = n4 << 2;
    const uint32_t tail  = n - start;
    RandomBitFlipFI_tail_kernel<<<dim3((tail + TPB - 1) / TPB), dim3(TPB), 0, stream>>>(
        (const uint32_t*)d_in[0], (uint32_t*)d_out, start, n);
  }
}